// GCN_62474594288248
// MI455X (gfx1250) — compile-verified
//
#include <hip/hip_runtime.h>
#include <math.h>

typedef __attribute__((ext_vector_type(2))) float v2f;
typedef __attribute__((ext_vector_type(8))) float v8f;

#define NNODES 50000

// ---------------- utility ----------------
__global__ void zero_f32(float* __restrict__ p, int n) {
  int i = blockIdx.x * blockDim.x + threadIdx.x;
  if (i < n) p[i] = 0.0f;
}

__global__ void deg_count(const int* __restrict__ dst, float* __restrict__ deg, int E) {
  int e = blockIdx.x * blockDim.x + threadIdx.x;
  if (e < E) atomicAdd(&deg[dst[e]], 1.0f);
}

__global__ void make_dinv(float* __restrict__ deg, int n) {
  int i = blockIdx.x * blockDim.x + threadIdx.x;
  if (i < n) deg[i] = rsqrtf(deg[i] + 1.0f);   // self loop => deg >= 1
}

// ---------------- fp32 WMMA GEMM: C[M,N] = A[M,K=128] @ B[128,N] ------------
// 256 threads = 8 waves; each wave owns a 16-row M tile and sweeps all N
// tiles, K unrolled as 32 x v_wmma_f32_16x16x4_f32 (exact fp32 products).
// B is staged in LDS *pair-packed*: ldsPair[(k/2)*N + col] = {B[k][col],
// B[k+1][col]} so every WMMA's B fragment is a single ds_load_b64
// (fragment needs K-pair {k0+2*hi, k0+2*hi+1} at one column).
template <int N>
__global__ void gemm_f32_wmma(const float* __restrict__ A, const float* __restrict__ B,
                              float* __restrict__ C, int M) {
  constexpr int K = 128;
  extern __shared__ float ldsRaw[];
  v2f* ldsPair = (v2f*)ldsRaw;

  const int tid = threadIdx.x;
  for (int i = tid; i < K * N; i += 256) {
    const int k = i / N, n = i - (i / N) * N;          // N is constexpr: shifts
    ldsRaw[((k >> 1) * N + n) * 2 + (k & 1)] = B[i];   // pair-packed store
  }
  __syncthreads();

  const int lane = tid & 31;
  const int wave = tid >> 5;
  const int hi   = lane >> 4;        // 0: lanes 0-15, 1: lanes 16-31
  const int lo   = lane & 15;
  const int m0   = blockIdx.x * 128 + wave * 16;
  if (m0 >= M) return;               // wave-uniform exit (after barrier)

  // A fragment: row striped over lanes 0-15; K pair {k0+2*hi, k0+2*hi+1}.
  int rowA = m0 + lo;
  if (rowA >= M) rowA = M - 1;       // clamp tail rows: EXEC all-1s at WMMA
  const float* aptr = A + (size_t)rowA * K + 2 * hi;

  for (int n0 = 0; n0 < N; n0 += 16) {
    v8f acc = {};
    const int col = n0 + lo;
#pragma unroll
    for (int k0 = 0; k0 < K; k0 += 4) {
      v2f a, b;
      a.x = aptr[k0 + 0];                              // A[row][k0 + 2*hi]
      a.y = aptr[k0 + 1];                              // A[row][k0 + 2*hi + 1]
      b = ldsPair[((k0 >> 1) + hi) * N + col];         // one ds_load_b64
      acc = __builtin_amdgcn_wmma_f32_16x16x4_f32(
          false, a, false, b, (short)0, acc, false, false);
    }
#pragma unroll
    for (int r = 0; r < 8; ++r) {
      const int row = m0 + r + 8 * hi;                 // C/D: VGPR r -> M = r + 8*hi
      if (row < M) C[(size_t)row * N + n0 + lo] = acc[r];
    }
  }
}

// ---------------- edge scatter: agg[dst] += h[src] * dinv[src]*dinv[dst] ----
// One wave per edge: coalesced 128B row reads, f32 atomics land in L2
// (h is L2-resident: 25.6MB << 192MB).
template <int D>
__global__ void scatter_edges(const float* __restrict__ h, const int* __restrict__ src,
                              const int* __restrict__ dst, const float* __restrict__ dinv,
                              float* __restrict__ agg, int E) {
  const int e = blockIdx.x * 8 + (threadIdx.x >> 5);
  if (e >= E) return;
  const int lane = threadIdx.x & 31;
  const int s = src[e], d = dst[e];
  const float norm = dinv[s] * dinv[d];
  const float* hs = h + (size_t)s * D;
  float* ad = agg + (size_t)d * D;
#pragma unroll
  for (int c = lane; c < D; c += 32) atomicAdd(&ad[c], hs[c] * norm);
}

// ---------------- layer-1 epilogue: self loop + bias + relu ----------------
template <int D>
__global__ void finalize_relu(float* __restrict__ agg, const float* __restrict__ h,
                              const float* __restrict__ dinv, const float* __restrict__ bias,
                              int n) {
  const int i = blockIdx.x * blockDim.x + threadIdx.x;
  if (i >= n * D) return;
  const int row = i / D, col = i & (D - 1);            // D constexpr pow2
  const float di = dinv[row];
  const float v = agg[i] + h[i] * (di * di) + bias[col];
  agg[i] = v > 0.0f ? v : 0.0f;
}

// ------- layer-2 epilogue: self loop + bias + log_softmax (row of 64) ------
// One wave per row; lane owns cols {lane, lane+32}; shuffle reductions.
__global__ void finalize_logsoftmax(float* __restrict__ out, const float* __restrict__ h2,
                                    const float* __restrict__ dinv, const float* __restrict__ bias,
                                    int n) {
  const int row = blockIdx.x * 8 + (threadIdx.x >> 5);
  if (row >= n) return;
  const int lane = threadIdx.x & 31;
  const float di = dinv[row];
  const float d2 = di * di;
  float* o = out + (size_t)row * 64;
  const float* h = h2 + (size_t)row * 64;

  float v0 = o[lane]      + h[lane]      * d2 + bias[lane];
  float v1 = o[lane + 32] + h[lane + 32] * d2 + bias[lane + 32];

  float m = fmaxf(v0, v1);
#pragma unroll
  for (int off = 16; off > 0; off >>= 1) m = fmaxf(m, __shfl_xor(m, off, 32));
  float s = expf(v0 - m) + expf(v1 - m);
#pragma unroll
  for (int off = 16; off > 0; off >>= 1) s += __shfl_xor(s, off, 32);
  const float lse = m + logf(s);
  o[lane]      = v0 - lse;
  o[lane + 32] = v1 - lse;
}

// ---------------- launch ----------------
extern "C" void kernel_launch(void* const* d_in, const int* in_sizes, int n_in,
                              void* d_out, int out_size, void* d_ws, size_t ws_size,
                              hipStream_t stream) {
  const float* x  = (const float*)d_in[0];
  const int*   ei = (const int*)  d_in[1];
  const float* W1 = (const float*)d_in[2];
  const float* b1 = (const float*)d_in[3];
  const float* W2 = (const float*)d_in[4];
  const float* b2 = (const float*)d_in[5];
  float* out = (float*)d_out;

  const int Nn = NNODES;
  const int E  = in_sizes[1] / 2;          // edge_index is (2, E) row-major
  const int* src = ei;
  const int* dst = ei + E;

  float* ws   = (float*)d_ws;
  float* dinv = ws;                               // Nn        (deg, then dinv in place)
  float* h1   = dinv + Nn;                        // Nn * 128
  float* a1   = h1 + (size_t)Nn * 128;            // Nn * 128
  float* h2   = a1 + (size_t)Nn * 128;            // Nn * 64

  // degree normalization
  zero_f32 <<<(Nn + 255) / 256, 256, 0, stream>>>(dinv, Nn);
  deg_count<<<(E  + 255) / 256, 256, 0, stream>>>(dst, dinv, E);
  make_dinv<<<(Nn + 255) / 256, 256, 0, stream>>>(dinv, Nn);

  // layer 1: h1 = x @ W1 ; agg + self-loop + bias + relu -> a1
  gemm_f32_wmma<128><<<(Nn + 127) / 128, 256, 128 * 128 * 4, stream>>>(x, W1, h1, Nn);
  zero_f32          <<<(Nn * 128 + 255) / 256, 256, 0, stream>>>(a1, Nn * 128);
  scatter_edges<128><<<(E + 7) / 8, 256, 0, stream>>>(h1, src, dst, dinv, a1, E);
  finalize_relu<128><<<(Nn * 128 + 255) / 256, 256, 0, stream>>>(a1, h1, dinv, b1, Nn);

  // layer 2: h2 = a1 @ W2 ; agg into d_out ; epilogue + log_softmax in place
  gemm_f32_wmma<64><<<(Nn + 127) / 128, 256, 128 * 64 * 4, stream>>>(a1, W2, h2, Nn);
  zero_f32         <<<(Nn * 64 + 255) / 256, 256, 0, stream>>>(out, Nn * 64);
  scatter_edges<64><<<(E + 7) / 8, 256, 0, stream>>>(h2, src, dst, dinv, out, E);
  finalize_logsoftmax<<<(Nn + 7) / 8, 256, 0, stream>>>(out, h2, dinv, b2, Nn);
}